// SModel_24756191494619
// MI455X (gfx1250) — compile-verified
//
#include <hip/hip_runtime.h>

typedef float v2f __attribute__((ext_vector_type(2)));
typedef float v8f __attribute__((ext_vector_type(8)));

#define N_S   50000
#define N_E   800000
#define FXS   128
#define MSGW  128
#define H2IN  705
#define H2PAD 708

__device__ __forceinline__ float lrelu(float x) { return x > 0.f ? x : 0.1f * x; }

__device__ __forceinline__ v8f wmma_f32_k4(v2f a, v2f b, v8f c) {
    return __builtin_amdgcn_wmma_f32_16x16x4_f32(false, a, false, b, (short)0, c, false, false);
}

// ---------------------------------------------------------------------------
// Edge pass: msg = leaky(in @ w1a + b1a) @ w1b + b1b, then scatter moments.
// PASS 1: acc0 += msg, acc1 += msg^2, count += 1
// PASS 2: d = msg - mean[src]; acc0 += d^3, acc1 += d^4
// 128 edges per block, 8 waves; wave w owns M-tile w (16 edges) x all 8 N-tiles.
// ---------------------------------------------------------------------------
template <int PASS>
__global__ __launch_bounds__(256) void edge_kernel(
    const float* __restrict__ x_t, const long long* __restrict__ edge_index,
    const float* __restrict__ edge_attr,
    const float* __restrict__ w1a, const float* __restrict__ b1a,
    const float* __restrict__ w1b, const float* __restrict__ b1b,
    const float* __restrict__ meanbuf,
    float* __restrict__ acc0, float* __restrict__ acc1, float* __restrict__ cnt)
{
    __shared__ float A[128 * 132];   // input rows, then activations (stride 132: bank-safe)
    __shared__ float W[128 * 132];   // w1a, then reloaded with w1b
    __shared__ int   srcI[128];

    const int tid   = threadIdx.x;
    const int lane  = tid & 31;
    const int wave  = tid >> 5;
    const int lm    = lane & 15;      // M or N within tile
    const int hl    = lane >> 4;      // K-half selector
    const long long eBase = (long long)blockIdx.x * 128;

    // stage w1a into LDS (coalesced)
    for (int i = tid; i < 128 * 128; i += 256)
        W[(i >> 7) * 132 + (i & 127)] = w1a[i];

    // stage concat(x_t[tgt], edge_attr) rows; 2 threads per edge row
    {
        const int r = tid >> 1, half = tid & 1;
        const long long e = eBase + r;
        const long long tg = edge_index[N_E + e];
        if (half == 0) srcI[r] = (int)edge_index[e];
        const float* sp = half ? (edge_attr + e * 64) : (x_t + tg * 64);
        float* dp = &A[r * 132 + half * 64];
        for (int c = 0; c < 64; ++c) dp[c] = sp[c];
    }
    __syncthreads();

    // ---- GEMM1: act = leaky(A @ w1a + b1a) ----
    v8f acc[8];
    for (int nt = 0; nt < 8; ++nt)
        for (int j = 0; j < 8; ++j) acc[nt][j] = 0.f;

    const int mrow = wave * 16 + lm;
    for (int kt = 0; kt < 32; ++kt) {
        const int k0 = kt * 4 + 2 * hl;
        v2f af; af.x = A[mrow * 132 + k0]; af.y = A[mrow * 132 + k0 + 1];
        for (int nt = 0; nt < 8; ++nt) {
            v2f bf;
            bf.x = W[(k0    ) * 132 + nt * 16 + lm];
            bf.y = W[(k0 + 1) * 132 + nt * 16 + lm];
            acc[nt] = wmma_f32_k4(af, bf, acc[nt]);
        }
    }
    __syncthreads();   // everyone done reading A and W

    // write activations into A; reload W with w1b
    for (int nt = 0; nt < 8; ++nt) {
        const int n = nt * 16 + lm;
        const float bias = b1a[n];
        for (int j = 0; j < 8; ++j) {
            const int m = j + 8 * hl;
            A[(wave * 16 + m) * 132 + n] = lrelu(acc[nt][j] + bias);
        }
    }
    for (int i = tid; i < 128 * 128; i += 256)
        W[(i >> 7) * 132 + (i & 127)] = w1b[i];
    __syncthreads();

    // ---- GEMM2: msg = act @ w1b + b1b ----
    for (int nt = 0; nt < 8; ++nt)
        for (int j = 0; j < 8; ++j) acc[nt][j] = 0.f;
    for (int kt = 0; kt < 32; ++kt) {
        const int k0 = kt * 4 + 2 * hl;
        v2f af; af.x = A[mrow * 132 + k0]; af.y = A[mrow * 132 + k0 + 1];
        for (int nt = 0; nt < 8; ++nt) {
            v2f bf;
            bf.x = W[(k0    ) * 132 + nt * 16 + lm];
            bf.y = W[(k0 + 1) * 132 + nt * 16 + lm];
            acc[nt] = wmma_f32_k4(af, bf, acc[nt]);
        }
    }

    // ---- scatter moments (L2-resident accumulators) ----
    for (int nt = 0; nt < 8; ++nt) {
        const int n = nt * 16 + lm;
        const float bias = b1b[n];
        for (int j = 0; j < 8; ++j) {
            const int m = j + 8 * hl;
            const float v = acc[nt][j] + bias;
            const long long s = srcI[wave * 16 + m];
            float* p0 = acc0 + s * MSGW + n;
            float* p1 = acc1 + s * MSGW + n;
            if (PASS == 1) {
                atomicAdd(p0, v);
                atomicAdd(p1, v * v);
            } else {
                const float d  = v - meanbuf[s * MSGW + n];
                const float d3 = d * d * d;
                atomicAdd(p0, d3);
                atomicAdd(p1, d3 * d);
            }
        }
    }
    if (PASS == 1 && tid < 128) atomicAdd(&cnt[srcI[tid]], 1.0f);
}

// ---------------------------------------------------------------------------
// Per-node stats: sum -> mean (in place), sumsq -> std (in place)
// ---------------------------------------------------------------------------
__global__ __launch_bounds__(256) void stats_kernel(
    float* __restrict__ sum, float* __restrict__ sumsq, const float* __restrict__ cnt)
{
    const long long i = (long long)blockIdx.x * 256 + threadIdx.x;
    if (i >= (long long)N_S * MSGW) return;
    const float c  = fmaxf(cnt[i >> 7], 1.0f);
    const float iv = 1.0f / c;
    const float mean = sum[i] * iv;
    const float m2   = sumsq[i] * iv;
    const float var  = fmaxf(m2 - mean * mean, 0.0f);
    sum[i]   = mean;
    sumsq[i] = sqrtf(var + 1e-6f);
}

// ---------------------------------------------------------------------------
// Node MLP: out = leaky(h @ w2a + b2a) @ w2b + b2b,  h = [x_s|count|mean|std|skew|kurt|u]
// 32 nodes per block; wave w: M-tile = w&1, N-tiles = (w>>1)*2 + {0,1}
// ---------------------------------------------------------------------------
__global__ __launch_bounds__(256) void node_kernel(
    const float* __restrict__ x_s, const float* __restrict__ u,
    const long long* __restrict__ batch_s,
    const float* __restrict__ meanb, const float* __restrict__ stdb,
    const float* __restrict__ sum3, const float* __restrict__ sum4,
    const float* __restrict__ cntb,
    const float* __restrict__ w2a, const float* __restrict__ b2a,
    const float* __restrict__ w2b, const float* __restrict__ b2b,
    float* __restrict__ out)
{
    __shared__ float H[32 * 716];    // h rows (708 used), stride 716: bank-safe
    __shared__ float WC[64 * 132];   // streamed weight chunk

    const int tid  = threadIdx.x;
    const int lane = tid & 31;
    const int wave = tid >> 5;
    const int lm   = lane & 15;
    const int hl   = lane >> 4;
    const int mt   = wave & 1;       // M-tile (16 nodes)
    const int ng   = wave >> 1;      // N-tile group (2 tiles)
    const long long nb = (long long)blockIdx.x * 32;

    // build h tile in LDS
    for (int idx = tid; idx < 32 * H2PAD; idx += 256) {
        const int r = idx / H2PAD, c = idx % H2PAD;
        long long node = nb + r;
        if (node >= N_S) node = N_S - 1;          // clamp; stores guarded later
        const long long b = node * MSGW;
        float v = 0.f;
        if (c < 128)       v = x_s[node * FXS + c];
        else if (c == 128) v = cntb[node];
        else if (c < 257)  v = meanb[b + (c - 129)];
        else if (c < 385)  v = stdb[b + (c - 257)];
        else if (c < 513) {
            const float s = stdb[b + (c - 385)];
            const float cn = fmaxf(cntb[node], 1.0f);
            v = (sum3[b + (c - 385)] / cn) / (s * s * s);
        } else if (c < 641) {
            const float s = stdb[b + (c - 513)];
            const float cn = fmaxf(cntb[node], 1.0f);
            const float s2 = s * s;
            v = (sum4[b + (c - 513)] / cn) / (s2 * s2);
        } else if (c < H2IN) {
            v = u[batch_s[node] * 64 + (c - 641)];
        }
        H[r * 716 + c] = v;
    }

    // ---- GEMM1 over K=708 in chunks of 64 ----
    v8f acc[2];
    for (int q = 0; q < 2; ++q)
        for (int j = 0; j < 8; ++j) acc[q][j] = 0.f;

    const int mrow = mt * 16 + lm;
    for (int chunk = 0; chunk < 12; ++chunk) {
        const int kbase = chunk * 64;
        const int klen  = (kbase + 64 <= H2PAD) ? 64 : (H2PAD - kbase);
        __syncthreads();
        for (int i = tid; i < klen * 128; i += 256) {
            const int k = i >> 7, n = i & 127;
            const int kg = kbase + k;
            WC[k * 132 + n] = (kg < H2IN) ? w2a[(long long)kg * 128 + n] : 0.f;
        }
        __syncthreads();
        for (int kt = 0; kt * 4 < klen; ++kt) {
            const int kl = kt * 4 + 2 * hl;          // local K in chunk
            const int kg = kbase + kl;               // global K in H
            v2f af; af.x = H[mrow * 716 + kg]; af.y = H[mrow * 716 + kg + 1];
            for (int q = 0; q < 2; ++q) {
                const int n = (ng * 2 + q) * 16 + lm;
                v2f bf;
                bf.x = WC[(kl    ) * 132 + n];
                bf.y = WC[(kl + 1) * 132 + n];
                acc[q] = wmma_f32_k4(af, bf, acc[q]);
            }
        }
    }
    __syncthreads();

    // hidden = leaky(acc + b2a) -> reuse H (stride 132)
    float* HID = H;
    for (int q = 0; q < 2; ++q) {
        const int n = (ng * 2 + q) * 16 + lm;
        const float bias = b2a[n];
        for (int j = 0; j < 8; ++j) {
            const int m = j + 8 * hl;
            HID[(mt * 16 + m) * 132 + n] = lrelu(acc[q][j] + bias);
        }
    }
    __syncthreads();

    // ---- GEMM2: K=128, w2b streamed in two 64-row chunks ----
    v8f acc2[2];
    for (int q = 0; q < 2; ++q)
        for (int j = 0; j < 8; ++j) acc2[q][j] = 0.f;
    for (int chunk = 0; chunk < 2; ++chunk) {
        const int kbase = chunk * 64;
        __syncthreads();
        for (int i = tid; i < 64 * 128; i += 256) {
            const int k = i >> 7, n = i & 127;
            WC[k * 132 + n] = w2b[(long long)(kbase + k) * 128 + n];
        }
        __syncthreads();
        for (int kt = 0; kt < 16; ++kt) {
            const int kl = kt * 4 + 2 * hl;
            v2f af; af.x = HID[mrow * 132 + kbase + kl]; af.y = HID[mrow * 132 + kbase + kl + 1];
            for (int q = 0; q < 2; ++q) {
                const int n = (ng * 2 + q) * 16 + lm;
                v2f bf;
                bf.x = WC[(kl    ) * 132 + n];
                bf.y = WC[(kl + 1) * 132 + n];
                acc2[q] = wmma_f32_k4(af, bf, acc2[q]);
            }
        }
    }

    // write out
    for (int q = 0; q < 2; ++q) {
        const int n = (ng * 2 + q) * 16 + lm;
        const float bias = b2b[n];
        for (int j = 0; j < 8; ++j) {
            const int m = j + 8 * hl;
            const long long node = nb + mt * 16 + m;
            if (node < N_S) out[node * FXS + n] = acc2[q][j] + bias;
        }
    }
}

// ---------------------------------------------------------------------------
extern "C" void kernel_launch(void* const* d_in, const int* in_sizes, int n_in,
                              void* d_out, int out_size, void* d_ws, size_t ws_size,
                              hipStream_t stream) {
    const float*     x_s     = (const float*)d_in[0];
    const float*     x_t     = (const float*)d_in[1];
    const long long* e_idx   = (const long long*)d_in[2];
    const float*     e_attr  = (const float*)d_in[3];
    const float*     u       = (const float*)d_in[4];
    const long long* batch_s = (const long long*)d_in[5];
    const float *w1a = (const float*)d_in[6],  *b1a = (const float*)d_in[7];
    const float *w1b = (const float*)d_in[8],  *b1b = (const float*)d_in[9];
    const float *w2a = (const float*)d_in[10], *b2a = (const float*)d_in[11];
    const float *w2b = (const float*)d_in[12], *b2b = (const float*)d_in[13];
    float* out = (float*)d_out;

    // workspace: sum | sumsq | sum3 | sum4 | count  (102.6 MB, L2-resident)
    float* sum   = (float*)d_ws;
    float* sumsq = sum   + (size_t)N_S * MSGW;
    float* sum3  = sumsq + (size_t)N_S * MSGW;
    float* sum4  = sum3  + (size_t)N_S * MSGW;
    float* cnt   = sum4  + (size_t)N_S * MSGW;

    hipMemsetAsync(d_ws, 0, ((size_t)4 * N_S * MSGW + N_S) * sizeof(float), stream);

    const int eblocks = N_E / 128;                       // 6250, exact
    edge_kernel<1><<<eblocks, 256, 0, stream>>>(x_t, e_idx, e_attr, w1a, b1a, w1b, b1b,
                                                sum, sum, sumsq, cnt);
    stats_kernel<<<(N_S * MSGW + 255) / 256, 256, 0, stream>>>(sum, sumsq, cnt);
    edge_kernel<2><<<eblocks, 256, 0, stream>>>(x_t, e_idx, e_attr, w1a, b1a, w1b, b1b,
                                                sum, sum3, sum4, cnt);
    node_kernel<<<(N_S + 31) / 32, 256, 0, stream>>>(x_s, u, batch_s, sum, sumsq, sum3,
                                                     sum4, cnt, w2a, b2a, w2b, b2b, out);
}